// QuantumDenseNet_67989332295725
// MI455X (gfx1250) — compile-verified
//
#include <hip/hip_runtime.h>
#include <hip/hip_bf16.h>

#define MODES   16
#define NPAIR   120          // 16*15/2
#define LINP    528          // 4*NPAIR + 3*MODES
#define LAYERS  6
#define OUTSZ   10

typedef __attribute__((ext_vector_type(2))) float v2f;
typedef __attribute__((ext_vector_type(8))) float v8f;

// ---------------------------------------------------------------------------
// Precompute kernel: one block (one wave32) per layer.
// Builds U1,U2 (16x16 complex) via the 120 beamsplitter rotations, forms the
// 32x32 symplectics, fuses W = S1^T * diag(scale) * S2^T, and emits W in the
// exact per-lane WMMA-B fragment layout plus the bias vector.
// ---------------------------------------------------------------------------
__global__ __launch_bounds__(32) void qdn_precompute(
    const float* __restrict__ lin, float2* __restrict__ wfrag,
    float* __restrict__ bias)
{
    const int l    = blockIdx.x;
    const int lane = threadIdx.x;
    const float* p = lin + l * LINP;

    __shared__ float Ur[16][16], Ui[16][16];
    __shared__ float S1[32][32], S2[32][32];
    __shared__ float W[32][32];
    __shared__ float scale[32];

    for (int pass = 0; pass < 2; ++pass) {
        const float* th = p + (pass == 0 ? 0     : 2 * NPAIR + MODES);
        const float* ph = p + (pass == 0 ? NPAIR : 3 * NPAIR + MODES);

        // U = I
        for (int row = 0; row < 16; ++row) {
            if (lane < 16) {
                Ur[row][lane] = (row == lane) ? 1.0f : 0.0f;
                Ui[row][lane] = 0.0f;
            }
        }
        __syncthreads();

        int k = 0;
        for (int i = 0; i < 15; ++i) {
            for (int j = i + 1; j < 16; ++j) {
                const float t  = th[k];
                const float f  = ph[k];
                ++k;
                const float c  = cosf(t), s  = sinf(t);
                const float er = cosf(f), ei = sinf(f);
                const int col = lane & 15;
                // read old rows i and j (all lanes)
                const float air = Ur[i][col], aii = Ui[i][col];
                const float ajr = Ur[j][col], aji = Ui[j][col];
                __syncthreads();
                if (lane < 16) {
                    // row_i' = c*a_i - conj(e)*s*a_j ; conj(e)=er - i*ei
                    Ur[i][col] = c * air - s * (er * ajr + ei * aji);
                    Ui[i][col] = c * aii - s * (er * aji - ei * ajr);
                } else {
                    // row_j' = e*s*a_i + c*a_j ; e=er + i*ei
                    Ur[j][col] = s * (er * air - ei * aii) + c * ajr;
                    Ui[j][col] = s * (er * aii + ei * air) + c * aji;
                }
                __syncthreads();
            }
        }

        // S = [[Re, -Im],[Im, Re]]
        float (*S)[32] = (pass == 0) ? S1 : S2;
        for (int row = 0; row < 32; ++row) {
            const int cc = lane & 15, rr = row & 15;
            float v;
            if (row < 16 && lane < 16)      v =  Ur[rr][cc];
            else if (row < 16)              v = -Ui[rr][cc];
            else if (lane < 16)             v =  Ui[rr][cc];
            else                            v =  Ur[rr][cc];
            S[row][lane] = v;
        }
        __syncthreads();
    }

    // scale = [exp(-r), exp(r)]
    {
        const float rv = p[2 * NPAIR + (lane & 15)];
        scale[lane] = (lane < 16) ? expf(-rv) : expf(rv);
    }
    __syncthreads();

    // W[a][b] = sum_k S1[k][a] * scale[k] * S2[b][k]   (lane owns column b)
    for (int a = 0; a < 32; ++a) {
        float acc = 0.0f;
        for (int k2 = 0; k2 < 32; ++k2)
            acc += S1[k2][a] * scale[k2] * S2[lane][k2];
        W[a][lane] = acc;
    }
    __syncthreads();

    // bias
    bias[l * 32 + lane] = p[4 * NPAIR + 2 * MODES + lane];

    // Emit B fragments: chunk c (K=4c..4c+3), tile t (N=16t..16t+15).
    // Main-kernel lane lm: hi=lm>>4, n=lm&15 holds {W[4c+2hi][N], W[4c+2hi+1][N]}.
    const int hi = lane >> 4, n = lane & 15;
    for (int c = 0; c < 8; ++c) {
        for (int t = 0; t < 2; ++t) {
            const int K0 = 4 * c + 2 * hi;
            const int N  = t * 16 + n;
            wfrag[((l * 8 + c) * 2 + t) * 32 + lane] =
                make_float2(W[K0][N], W[K0 + 1][N]);
        }
    }
}

// ---------------------------------------------------------------------------
// Main batched kernel: one wave32 per 16-sample tile, 8 waves per block.
// Layer 0: 8 WMMAs (zero p-half skipped); layers 1..5: 16 WMMAs each.
// Inter-layer C-layout -> A-layout transpose through a wave-private 2 KB LDS
// slice; same-wave LDS ops are processed in order (ISA: "LDS ... in order"),
// so a s_wait_dscnt 0 fence suffices — no block-wide barrier needed.
// ---------------------------------------------------------------------------
__global__ __launch_bounds__(256) void qdn_main(
    const float* __restrict__ batch,
    const float* __restrict__ actp,
    const float* __restrict__ lastp,
    const float2* __restrict__ wfrag,
    const float* __restrict__ bias,
    float* __restrict__ out,
    int nsamples)
{
    __shared__ float smem[8][16][32];   // 16 KB: per-wave 16x32 staging tile

    const int lane  = threadIdx.x & 31;
    const int w     = threadIdx.x >> 5;
    const int hi    = lane >> 4;
    const int n     = lane & 15;
    const int tile  = blockIdx.x * 8 + w;
    const int sbase = tile * 16;
    if (sbase >= nsamples) return;      // wave-uniform exit; no partial tiles

    v2f a[8];

    // Layer-0 A operand straight from global; K=16..31 (p-half) is all zero.
    #pragma unroll
    for (int c = 0; c < 4; ++c) {
        const float* src = batch + (size_t)(sbase + n) * 16 + 4 * c + 2 * hi;
        a[c] = *(const v2f*)src;        // global_load_b64
    }

    v8f c0, c1;                          // x-modes tile / p-modes tile

    for (int l = 0; l < LAYERS; ++l) {
        if (l > 0) {
            // C-layout (row per VGPR) -> A-layout (row per lane) via LDS
            #pragma unroll
            for (int r = 0; r < 8; ++r) {
                smem[w][r + 8 * hi][n]      = c0[r];
                smem[w][r + 8 * hi][16 + n] = c1[r];
            }
            // Wave-local fence: staging stores retired before transposed loads.
            asm volatile("s_wait_dscnt 0x0" ::: "memory");
            #pragma unroll
            for (int c = 0; c < 8; ++c) {
                const float* src = &smem[w][n][4 * c + 2 * hi];
                a[c] = *(const v2f*)src; // ds_load_b64
            }
        }

        // Seed accumulators with the bias (every row gets d[N])
        const float b0 = bias[l * 32 + n];
        const float b1 = bias[l * 32 + 16 + n];
        #pragma unroll
        for (int r = 0; r < 8; ++r) { c0[r] = b0; c1[r] = b1; }

        const int nchunks = (l == 0) ? 4 : 8;   // layer 0: zero p-half skipped
        #pragma unroll
        for (int c = 0; c < 8; ++c) {
            if (c >= nchunks) break;
            const v2f bf0 = *(const v2f*)&wfrag[((l * 8 + c) * 2 + 0) * 32 + lane];
            const v2f bf1 = *(const v2f*)&wfrag[((l * 8 + c) * 2 + 1) * 32 + lane];
            c0 = __builtin_amdgcn_wmma_f32_16x16x4_f32(
                     false, a[c], false, bf0, (short)0, c0, false, false);
            c1 = __builtin_amdgcn_wmma_f32_16x16x4_f32(
                     false, a[c], false, bf1, (short)0, c1, false, false);
        }

        // Kerr activation: lane n holds mode n for both x (c0) and p (c1).
        // Last layer: mask modes >= OUTSZ by kappa = 0 (identity rotation).
        float kap;
        if (l < LAYERS - 1) kap = actp[l * MODES + n];
        else                kap = (n < OUTSZ) ? lastp[n] : 0.0f;

        #pragma unroll
        for (int r = 0; r < 8; ++r) {
            const float x  = c0[r];
            const float pq = c1[r];
            const float ang = kap * (x * x + pq * pq);
            const float cc = __cosf(ang);
            const float ss = __sinf(ang);
            c0[r] =  cc * x + ss * pq;
            c1[r] = -ss * x + cc * pq;
        }
    }

    // out[:, :10] = x-modes 0..9 ; lane n owns column n, VGPR r owns row r+8*hi
    if (n < OUTSZ) {
        #pragma unroll
        for (int r = 0; r < 8; ++r)
            out[(size_t)(sbase + r + 8 * hi) * OUTSZ + n] = c0[r];
    }
}

// ---------------------------------------------------------------------------
extern "C" void kernel_launch(void* const* d_in, const int* in_sizes, int n_in,
                              void* d_out, int out_size, void* d_ws, size_t ws_size,
                              hipStream_t stream) {
    const float* batch = (const float*)d_in[0];   // (B, 16) f32
    const float* lin   = (const float*)d_in[1];   // (6, 528) f32
    const float* actp  = (const float*)d_in[2];   // (5, 16) f32
    const float* lastp = (const float*)d_in[3];   // (16,) f32
    float* out = (float*)d_out;                   // (B, 10) f32

    float2* wfrag = (float2*)d_ws;                // 6*8*2*32 float2 = 24 KB
    float*  bias  = (float*)((char*)d_ws + (size_t)LAYERS * 8 * 2 * 32 * sizeof(float2));

    const int nsamples = in_sizes[0] / MODES;     // 524288
    const int tiles    = nsamples / 16;           // 32768 waves
    const int blocks   = (tiles + 7) / 8;         // 8 waves / block

    qdn_precompute<<<LAYERS, 32, 0, stream>>>(lin, wfrag, bias);
    qdn_main<<<blocks, 256, 0, stream>>>(batch, actp, lastp, wfrag, bias, out, nsamples);
}